// Slater_21285857919324
// MI455X (gfx1250) — compile-verified
//
#include <hip/hip_runtime.h>

typedef __attribute__((ext_vector_type(2))) float v2f;
typedef __attribute__((ext_vector_type(4))) float v4f;
typedef __attribute__((ext_vector_type(8))) float v8f;
typedef __attribute__((ext_vector_type(4))) int   v4i;

#define NBLK 1568
#define NTHR 256
#define PACKED_OFF 16384  // byte offset of packed atom records in d_ws

// --- prologue: invert 3x3 box into ws[0..8]; pack {x,y,z,type} per atom ---
__global__ __launch_bounds__(256) void slater_pack_kernel(
    const float* __restrict__ coords,
    const int*   __restrict__ types32,  // int64 viewed as int32 pairs (LE low dword)
    const float* __restrict__ box,
    float*       __restrict__ ws,
    v4f*         __restrict__ packed,   // may be null when ws too small
    int N, int do_pack)
{
    int t = blockIdx.x * 256 + threadIdx.x;
    if (t == 0) {
        float b00 = box[0], b01 = box[1], b02 = box[2];
        float b10 = box[3], b11 = box[4], b12 = box[5];
        float b20 = box[6], b21 = box[7], b22 = box[8];
        float c00 =  (b11 * b22 - b12 * b21);
        float c01 = -(b10 * b22 - b12 * b20);
        float c02 =  (b10 * b21 - b11 * b20);
        float det = b00 * c00 + b01 * c01 + b02 * c02;
        float id  = 1.0f / det;
        ws[0] = c00 * id;
        ws[1] = (b02 * b21 - b01 * b22) * id;
        ws[2] = (b01 * b12 - b02 * b11) * id;
        ws[3] = c01 * id;
        ws[4] = (b00 * b22 - b02 * b20) * id;
        ws[5] = (b02 * b10 - b00 * b12) * id;
        ws[6] = c02 * id;
        ws[7] = (b01 * b20 - b00 * b21) * id;
        ws[8] = (b00 * b11 - b01 * b10) * id;
    }
    if (do_pack && t < N) {
        v4f rec;
        rec[0] = coords[3 * t + 0];
        rec[1] = coords[3 * t + 1];
        rec[2] = coords[3 * t + 2];
        rec[3] = __int_as_float(types32[2 * t]);
        packed[t] = rec;
    }
}

// Wave32 sum via two chained V_WMMA_F32_16X16X4_F32 (no shuffles, no LDS):
//  WMMA1: A=acc(K0/K2), B=ones -> D1[m,n] = acc[m]+acc[m+16] = rowsum(m)
//  VALU:  h = sum of 8 D1 regs  -> lanes 0-15: S_lo = rows 0-7; lanes 16-31: S_hi
//  WMMA2: A=h, B=ones -> D2[m,n] = S_lo + S_hi = wave total (all lanes)
__device__ __forceinline__ float wave_reduce_wmma(float acc) {
    v2f ones; ones[0] = 1.0f; ones[1] = 1.0f;
    v2f a1;   a1[0] = acc;    a1[1] = 0.0f;
    v8f c = {};
    v8f d1 = __builtin_amdgcn_wmma_f32_16x16x4_f32(false, a1, false, ones,
                                                   (short)0, c, false, false);
    float h = ((d1[0] + d1[1]) + (d1[2] + d1[3])) +
              ((d1[4] + d1[5]) + (d1[6] + d1[7]));
    v2f a2; a2[0] = h; a2[1] = 0.0f;
    v8f d2 = __builtin_amdgcn_wmma_f32_16x16x4_f32(false, a2, false, ones,
                                                   (short)0, c, false, false);
    return d2[0];
}

template <bool USE_PACKED>
__global__ __launch_bounds__(NTHR) void slater_main_kernel(
    const float* __restrict__ coords,
    const int*   __restrict__ pairs32,   // int64 pairs viewed as int32[4]
    const float* __restrict__ box,
    const float* __restrict__ Ag,
    const float* __restrict__ Bg,
    const int*   __restrict__ cutoffp,
    const int*   __restrict__ types32,
    const float* __restrict__ invb,      // ws[0..8]
    const v4f*   __restrict__ packed,    // {x,y,z,type} per atom
    float*       __restrict__ partials,  // [NBLK]
    int P)
{
    __shared__ float sA[256];
    __shared__ float sB[256];
    __shared__ float wsum[NTHR / 32];

    const int tid = threadIdx.x;
    sA[tid] = Ag[tid];
    sB[tid] = Bg[tid];

    const float i00 = invb[0], i01 = invb[1], i02 = invb[2];
    const float i10 = invb[3], i11 = invb[4], i12 = invb[5];
    const float i20 = invb[6], i21 = invb[7], i22 = invb[8];
    const float h00 = box[0],  h01 = box[1],  h02 = box[2];
    const float h10 = box[3],  h11 = box[4],  h12 = box[5];
    const float h20 = box[6],  h21 = box[7],  h22 = box[8];
    const float cutf = (float)cutoffp[0];
    __syncthreads();

    float acc = 0.0f;
    const int stride = NBLK * NTHR;
    for (int p = blockIdx.x * NTHR + tid; p < P; p += stride) {
        v4i pr = __builtin_nontemporal_load((const v4i*)pairs32 + p);
        int i = pr[0];
        int j = pr[2];

        float dx, dy, dz;
        int ti, tj;
        if (USE_PACKED) {
            // one aligned 16B gather per atom: {x,y,z,type}
            v4f ri = packed[i];
            v4f rj = packed[j];
            dx = rj[0] - ri[0];
            dy = rj[1] - ri[1];
            dz = rj[2] - ri[2];
            ti = __float_as_int(ri[3]);
            tj = __float_as_int(rj[3]);
        } else {
            dx = coords[3 * j + 0] - coords[3 * i + 0];
            dy = coords[3 * j + 1] - coords[3 * i + 1];
            dz = coords[3 * j + 2] - coords[3 * i + 2];
            ti = types32[2 * i];
            tj = types32[2 * j];
        }

        float s0 = dx * i00 + dy * i10 + dz * i20;
        float s1 = dx * i01 + dy * i11 + dz * i21;
        float s2 = dx * i02 + dy * i12 + dz * i22;
        s0 -= rintf(s0);
        s1 -= rintf(s1);
        s2 -= rintf(s2);
        float r0 = s0 * h00 + s1 * h10 + s2 * h20;
        float r1 = s0 * h01 + s1 * h11 + s2 * h21;
        float r2 = s0 * h02 + s1 * h12 + s2 * h22;
        float r  = sqrtf(r0 * r0 + r1 * r1 + r2 * r2);

        int idx = (ti << 4) + tj;
        float x  = sB[idx] * r;
        float pf = x * x * (1.0f / 3.0f) + x + 1.0f;
        float e  = sA[idx] * pf * __expf(-x);
        acc += (r <= cutf) ? e : 0.0f;
    }

    // in-wave reduction: two chained WMMAs (EXEC all-ones: converged, full blocks)
    float s = wave_reduce_wmma(acc);

    const int wave = tid >> 5;
    if ((tid & 31) == 0) wsum[wave] = s;
    __syncthreads();
    if (tid == 0) {
        float bs = 0.0f;
        #pragma unroll
        for (int w = 0; w < NTHR / 32; ++w) bs += wsum[w];
        partials[blockIdx.x] = bs;
    }
}

// --- deterministic final tree reduction of block partials ---
__global__ __launch_bounds__(1024) void slater_reduce_kernel(
    const float* __restrict__ partials, int n, float* __restrict__ out)
{
    __shared__ float sm[1024];
    float s = 0.0f;
    for (int i = threadIdx.x; i < n; i += 1024) s += partials[i];
    sm[threadIdx.x] = s;
    __syncthreads();
    for (int off = 512; off >= 1; off >>= 1) {
        if (threadIdx.x < (unsigned)off) sm[threadIdx.x] += sm[threadIdx.x + off];
        __syncthreads();
    }
    if (threadIdx.x == 0) out[0] = sm[0];
}

extern "C" void kernel_launch(void* const* d_in, const int* in_sizes, int n_in,
                              void* d_out, int out_size, void* d_ws, size_t ws_size,
                              hipStream_t stream) {
    const float* coords = (const float*)d_in[0];
    const int*   pairs  = (const int*)  d_in[1];
    const float* box    = (const float*)d_in[2];
    const float* A      = (const float*)d_in[3];
    const float* B      = (const float*)d_in[4];
    const int*   cutoff = (const int*)  d_in[5];
    const int*   types  = (const int*)  d_in[6];

    const int P = in_sizes[1] / 2;
    const int N = in_sizes[0] / 3;

    float* ws       = (float*)d_ws;
    float* partials = ws + 64;  // ws[0..8] = inv(box); [64..64+NBLK) = partials
    v4f*   packed   = (v4f*)((char*)d_ws + PACKED_OFF);

    const bool use_packed =
        ws_size >= (size_t)PACKED_OFF + (size_t)N * sizeof(v4f);

    slater_pack_kernel<<<(N + 255) / 256, 256, 0, stream>>>(
        coords, types, box, ws, use_packed ? packed : nullptr, N,
        use_packed ? 1 : 0);

    if (use_packed) {
        slater_main_kernel<true><<<NBLK, NTHR, 0, stream>>>(
            coords, pairs, box, A, B, cutoff, types, ws, packed, partials, P);
    } else {
        slater_main_kernel<false><<<NBLK, NTHR, 0, stream>>>(
            coords, pairs, box, A, B, cutoff, types, ws, packed, partials, P);
    }
    slater_reduce_kernel<<<1, 1024, 0, stream>>>(partials, NBLK, (float*)d_out);
}